// SmoothUpsample_3951369912804
// MI455X (gfx1250) — compile-verified
//
#include <hip/hip_runtime.h>
#include <stdint.h>

// ---------------- configuration ----------------
#define C_DIM            64
#define FACES_PER_BLOCK  64          // faces staged per block (TDM tiles)
#define BLK              256         // 8 wave32 per block
#define SCAN_ITEMS       16
#define SCAN_CHUNK       (BLK * SCAN_ITEMS)   // 4096 pad-flags per scan block

typedef __attribute__((ext_vector_type(4))) unsigned int v4u_t;
typedef __attribute__((ext_vector_type(8))) int          v8i_t;
typedef __attribute__((ext_vector_type(4))) int          v4i_t;

// Issue one 1-D TDM transfer (contiguous tile, tail-clamped so no OOB).
// log2sz: 0=1B,1=2B,2=4B,3=8B elements. elems must fit in 16 bits.
__device__ __forceinline__ void tdm_1d(unsigned long long ga, unsigned int lds,
                                       unsigned int elems, unsigned int log2sz,
                                       bool is_store)
{
    // D# group 0: count=1 | lds_addr | global_addr(57b) | type=2
    v4u_t g0;
    g0[0] = 1u;
    g0[1] = lds;
    g0[2] = (unsigned int)(ga & 0xFFFFFFFFull);
    g0[3] = (unsigned int)((ga >> 32) & 0x1FFFFFFull) | (2u << 30);
    // D# group 1: data_size; tensor_dim0=elems; tensor_dim1=1; tile_dim0=elems
    v8i_t g1;
    g1[0] = (int)(log2sz << 16);                                // data_size @17:16
    g1[1] = (int)((elems & 0xFFFFu) << 16);                     // tensor_dim0[15:0] @63:48
    g1[2] = (int)(((elems >> 16) & 0xFFFFu) | (1u << 16));      // tensor_dim0[31:16] @79:64, tensor_dim1=1 @95:80
    g1[3] = (int)((elems & 0xFFFFu) << 16);                     // tile_dim0 @127:112
    g1[4] = 0;                                                  // tile_dim1=0, tile_dim2=0
    g1[5] = (int)elems;                                         // tensor_dim0_stride[31:0]
    g1[6] = 0;
    g1[7] = 0;
    v4i_t z; z[0] = 0; z[1] = 0; z[2] = 0; z[3] = 0;
#if defined(__clang_major__) && (__clang_major__ >= 23)
    v8i_t z8; z8[0]=0; z8[1]=0; z8[2]=0; z8[3]=0; z8[4]=0; z8[5]=0; z8[6]=0; z8[7]=0;
    if (is_store) __builtin_amdgcn_tensor_store_from_lds(g0, g1, z, z, z8, 0);
    else          __builtin_amdgcn_tensor_load_to_lds (g0, g1, z, z, z8, 0);
#else
    if (is_store) __builtin_amdgcn_tensor_store_from_lds(g0, g1, z, z, 0);
    else          __builtin_amdgcn_tensor_load_to_lds (g0, g1, z, z, 0);
#endif
}

__device__ __forceinline__ unsigned int byte_sum01(unsigned int w)
{
    // bytes are 0/1 -> sum of 4 bytes lands in the top byte, no carries
    return (w * 0x01010101u) >> 24;
}

// ---------------- pass 1: per-block count of non-pad slots ----------------
__global__ void __launch_bounds__(BLK) su_count_kernel(
    const unsigned char* __restrict__ is_pad, int* __restrict__ bsums, int n)
{
    __shared__ int s[BLK];
    const int tid = threadIdx.x;
    const long long base = (long long)blockIdx.x * SCAN_CHUNK + tid * SCAN_ITEMS;
    int c = 0;
    if (base + SCAN_ITEMS <= (long long)n) {
        const uint4 v = *(const uint4*)(is_pad + base);   // one b128 load
        unsigned pads = byte_sum01(v.x) + byte_sum01(v.y)
                      + byte_sum01(v.z) + byte_sum01(v.w);
        c = SCAN_ITEMS - (int)pads;
    } else {
        for (int i = 0; i < SCAN_ITEMS; ++i) {
            long long j = base + i;
            if (j < (long long)n) c += (is_pad[j] == 0);
        }
    }
    s[tid] = c;
    __syncthreads();
    for (int off = BLK >> 1; off > 0; off >>= 1) {
        if (tid < off) s[tid] += s[tid + off];
        __syncthreads();
    }
    if (tid == 0) bsums[blockIdx.x] = s[0];
}

// ---------------- pass 2: exclusive scan of block sums (tiny) ----------------
__global__ void su_scan_sums_kernel(int* __restrict__ bsums, int nb)
{
    if (threadIdx.x == 0 && blockIdx.x == 0) {
        int run = 0;
        for (int i = 0; i < nb; ++i) { int v = bsums[i]; bsums[i] = run; run += v; }
    }
}

// ---------------- pass 3: remap[j] = pad ? -1 : (int)pool_map[rank[j]] ----------------
__global__ void __launch_bounds__(BLK) su_remap_kernel(
    const unsigned char* __restrict__ is_pad,
    const long long*     __restrict__ pool_map,
    const int*           __restrict__ bsums,
    int*                 __restrict__ remap,
    int n, int n_fine)
{
    __shared__ int s[BLK];
    const int tid = threadIdx.x;
    const long long base = (long long)blockIdx.x * SCAN_CHUNK + tid * SCAN_ITEMS;

    unsigned int w[4] = {0xFFFFFFFFu, 0xFFFFFFFFu, 0xFFFFFFFFu, 0xFFFFFFFFu};
    const bool full = (base + SCAN_ITEMS <= (long long)n);
    if (full) {
        const uint4 v = *(const uint4*)(is_pad + base);
        w[0] = v.x; w[1] = v.y; w[2] = v.z; w[3] = v.w;
    } else {
        for (int i = 0; i < SCAN_ITEMS; ++i) {
            long long j = base + i;
            unsigned char b = (j < (long long)n) ? is_pad[j] : 1;  // OOB -> "pad"
            w[i >> 2] = (w[i >> 2] & ~(0xFFu << ((i & 3) * 8)))
                      | ((unsigned int)b << ((i & 3) * 8));
        }
    }
    int c = SCAN_ITEMS - (int)(byte_sum01(w[0] & 0x01010101u) + byte_sum01(w[1] & 0x01010101u)
                             + byte_sum01(w[2] & 0x01010101u) + byte_sum01(w[3] & 0x01010101u));
    s[tid] = c;
    __syncthreads();
    // Hillis-Steele inclusive scan over the 256 per-thread counts
    for (int off = 1; off < BLK; off <<= 1) {
        int add = (tid >= off) ? s[tid - off] : 0;
        __syncthreads();
        s[tid] += add;
        __syncthreads();
    }
    int rank = bsums[blockIdx.x] + ((tid == 0) ? 0 : s[tid - 1]);
    for (int i = 0; i < SCAN_ITEMS; ++i) {
        long long j = base + i;
        if (j < (long long)n) {
            const unsigned int pad = (w[i >> 2] >> ((i & 3) * 8)) & 0x01u;
            if (!pad) {
                int r = (rank < n_fine) ? rank : (n_fine - 1);   // matches jnp.clip
                remap[j] = (int)pool_map[r];
                rank++;
            } else {
                remap[j] = -1;
            }
        }
    }
}

// ---------------- main kernel: TDM-staged 9-tap gather blur ----------------
__global__ void __launch_bounds__(BLK) su_blur_kernel(
    const float*     __restrict__ x,       // [N_coarse, 64]
    const long long* __restrict__ nbr,     // [N_fine, 9]
    const int*       __restrict__ remap,   // [pad_size]
    float*           __restrict__ out,     // [N_fine, 64]
    int n_fine)
{
    __shared__ __align__(16) long long s_nbr[FACES_PER_BLOCK * 9];     // 4.5 KB
    __shared__ int                     s_m  [FACES_PER_BLOCK * 9];     // 2.25 KB
    __shared__ __align__(16) float     s_out[FACES_PER_BLOCK * C_DIM]; // 16 KB

    const int f0 = blockIdx.x * FACES_PER_BLOCK;
    int nf = n_fine - f0;
    if (nf > FACES_PER_BLOCK) nf = FACES_PER_BLOCK;
    const int elems = nf * 9;                       // int64 elements, <= 576

    // Wave 0: TDM-load this block's contiguous face_neighborhood slice to LDS.
    if (threadIdx.x < 32) {
        tdm_1d((unsigned long long)(uintptr_t)(nbr + (size_t)f0 * 9),
               (unsigned int)(unsigned long long)(uintptr_t)(&s_nbr[0]),
               (unsigned int)elems, 3u, false);
        __builtin_amdgcn_s_wait_tensorcnt(0);
    }
    __syncthreads();

    // Cooperative pass: resolve remap once per (face,tap), share via LDS.
    for (int e = threadIdx.x; e < elems; e += BLK)
        s_m[e] = remap[s_nbr[e]];
    __syncthreads();

    const int lane = threadIdx.x & 15;   // 16 lanes/face, float4 each -> 64 ch
    const int grp  = threadIdx.x >> 4;   // 16 face-groups per block
    const int coff = lane * 4;

    const float w0 = 0.25f, w1 = 0.125f, w2 = 0.0625f;
    const float W[9] = { w0, w1, w2, w1, w2, w1, w2, w1, w2 };

#pragma unroll
    for (int r = 0; r < FACES_PER_BLOCK; r += 16) {
        const int fl = r + grp;
        if (fl < nf) {
            float ax = 0.f, ay = 0.f, az = 0.f, aw = 0.f, corr = 0.f;
            // Branchless taps: pad neighbors (m < 0) read row 0 with weight 0,
            // so all 9 b128 gathers issue unconditionally and overlap.
#pragma unroll
            for (int t = 0; t < 9; ++t) {
                const int m   = s_m[fl * 9 + t];          // -1 on pad slot
                const float w = (m >= 0) ? W[t] : 0.0f;
                const int row = (m >= 0) ? m : 0;
                const float4 v = *(const float4*)(x + (size_t)row * C_DIM + coff);
                ax = fmaf(w, v.x, ax); ay = fmaf(w, v.y, ay);
                az = fmaf(w, v.z, az); aw = fmaf(w, v.w, aw);
                corr += w;
            }
            float4 o;
            o.x = ax / corr; o.y = ay / corr; o.z = az / corr; o.w = aw / corr;
            *(float4*)(&s_out[fl * C_DIM + coff]) = o;
        }
    }
    __syncthreads();

    // Wave 0: TDM-store the finished [nf x 64] f32 tile back to global.
    if (threadIdx.x < 32) {
        tdm_1d((unsigned long long)(uintptr_t)(out + (size_t)f0 * C_DIM),
               (unsigned int)(unsigned long long)(uintptr_t)(&s_out[0]),
               (unsigned int)(nf * C_DIM), 2u, true);
        __builtin_amdgcn_s_wait_tensorcnt(0);
    }
}

// ---------------- launcher ----------------
extern "C" void kernel_launch(void* const* d_in, const int* in_sizes, int n_in,
                              void* d_out, int out_size, void* d_ws, size_t ws_size,
                              hipStream_t stream)
{
    const float*         x        = (const float*)d_in[0];
    const long long*     nbr      = (const long long*)d_in[1];
    const unsigned char* is_pad   = (const unsigned char*)d_in[2];
    // d_in[3] = pad_size scalar (derive from in_sizes instead)
    const long long*     pool_map = (const long long*)d_in[4];

    const int pad_n  = in_sizes[2];
    const int n_fine = in_sizes[4];

    // Workspace layout: remap[pad_n] (int32), then block sums.
    int*   remap = (int*)d_ws;
    size_t rb    = (((size_t)pad_n * sizeof(int)) + 255) & ~(size_t)255;
    int*   bsums = (int*)((char*)d_ws + rb);
    const int nb = (pad_n + SCAN_CHUNK - 1) / SCAN_CHUNK;

    hipLaunchKernelGGL(su_count_kernel, dim3(nb), dim3(BLK), 0, stream,
                       is_pad, bsums, pad_n);
    hipLaunchKernelGGL(su_scan_sums_kernel, dim3(1), dim3(1), 0, stream,
                       bsums, nb);
    hipLaunchKernelGGL(su_remap_kernel, dim3(nb), dim3(BLK), 0, stream,
                       is_pad, pool_map, bsums, remap, pad_n, n_fine);

    const int gb = (n_fine + FACES_PER_BLOCK - 1) / FACES_PER_BLOCK;
    hipLaunchKernelGGL(su_blur_kernel, dim3(gb), dim3(BLK), 0, stream,
                       x, nbr, remap, (float*)d_out, n_fine);
}